// Edge2NodeAttnLayer_8735963480243
// MI455X (gfx1250) — compile-verified
//
#include <hip/hip_runtime.h>
#include <math.h>
#include <stdint.h>

// ---------------- problem constants ----------------
#define BB 4
#define NN 192
#define HN 256
#define HE 128
#define NH 8
#define DH 32
#define EE 18336            // N*(N-1)/2
#define ROWS_NODE (BB * NN)       // 768
#define ROWS_EDGE (BB * EE)       // 73344

typedef float v2f __attribute__((ext_vector_type(2)));
typedef float v8f __attribute__((ext_vector_type(8)));

#define V8F_ZERO {0.f,0.f,0.f,0.f,0.f,0.f,0.f,0.f}

// ============================================================
// Kernel 1: node LayerNorm  (one block per row, 256 threads)
// ============================================================
__global__ __launch_bounds__(256)
void node_ln_kernel(const float* __restrict__ x, const float* __restrict__ g,
                    const float* __restrict__ bvec, float* __restrict__ y) {
  __shared__ float ps[8], ps2[8], smv[2];
  const int tid = threadIdx.x;
  const size_t row = blockIdx.x;
  const float v = x[row * HN + tid];
  float s = v, s2 = v * v;
  for (int o = 16; o; o >>= 1) {
    s  += __shfl_xor(s,  o, 32);
    s2 += __shfl_xor(s2, o, 32);
  }
  if ((tid & 31) == 0) { ps[tid >> 5] = s; ps2[tid >> 5] = s2; }
  __syncthreads();
  if (tid == 0) {
    float ts = 0.f, ts2 = 0.f;
    for (int i = 0; i < 8; ++i) { ts += ps[i]; ts2 += ps2[i]; }
    const float mean = ts * (1.f / HN);
    const float var  = ts2 * (1.f / HN) - mean * mean;
    smv[0] = mean;
    smv[1] = rsqrtf(var + 1e-5f);
  }
  __syncthreads();
  y[row * HN + tid] = (v - smv[0]) * smv[1] * g[tid] + bvec[tid];
}

// ============================================================
// Kernel 2: generic f32 WMMA GEMM  out(M,Nc) = A(M,K) @ W(K,Nc) + bias
// one 16x16 output tile per wave, K stepped by 4 (V_WMMA_F32_16X16X4_F32)
// ============================================================
__global__ __launch_bounds__(256)
void gemm_f32_wmma(const float* __restrict__ A, const float* __restrict__ W,
                   const float* __restrict__ bias, float* __restrict__ out,
                   int M, int K, int Nc) {
  const int ntiles_n = Nc >> 4;
  const int total    = (M >> 4) * ntiles_n;
  const int wave = threadIdx.x >> 5;
  const int lane = threadIdx.x & 31;
  const int tile = blockIdx.x * (blockDim.x >> 5) + wave;
  if (tile >= total) return;                       // wave-uniform exit
  const int tileM = tile / ntiles_n;
  const int tileN = tile - tileM * ntiles_n;
  const int half = lane >> 4;
  const int l    = lane & 15;
  const int arow = tileM * 16 + l;
  const int bcol = tileN * 16 + l;
  const float* Arow = A + (size_t)arow * K;

  v8f acc = V8F_ZERO;
  for (int k = 0; k < K; k += 4) {
    const int kb = k + half * 2;
    v2f a = *(const v2f*)(Arow + kb);              // A[arow][kb..kb+1]
    v2f b;
    b.x = W[(size_t)kb * Nc + bcol];
    b.y = W[(size_t)(kb + 1) * Nc + bcol];
    acc = __builtin_amdgcn_wmma_f32_16x16x4_f32(false, a, false, b,
                                                (short)0, acc, false, false);
  }
  const int cc = tileN * 16 + l;
  const float bv = bias[cc];
  for (int r = 0; r < 8; ++r) {
    const int rr = tileM * 16 + (half ? 8 + r : r);
    out[(size_t)rr * Nc + cc] = acc[r] + bv;
  }
}

// ============================================================
// Kernel 3: fused edge LayerNorm + (ei @ W_edge)*mask + sigmoid-logits
// 64 edge rows per block, 256 threads (8 waves).
// W_edge consumed as 16x256 K-panels staged global->LDS with
// GLOBAL_LOAD_ASYNC_TO_LDS_B128 (ASYNCcnt), WMMA fed from LDS only.
// ============================================================
__global__ __launch_bounds__(256)
void edge_ln_gemm(const float* __restrict__ edge_feat, const float* __restrict__ mask,
                  const float* __restrict__ lg, const float* __restrict__ lb,
                  const float* __restrict__ W_edge, const float* __restrict__ b_edge,
                  const float* __restrict__ W_logit, const float* __restrict__ b_logit,
                  float* __restrict__ edge_new, float* __restrict__ sig_out) {
  __shared__ float ei[64][HE];        // 32 KB normalized edge rows
  __shared__ float wpan[16][HN];      // 16 KB W_edge K-panel (async-staged)
  __shared__ float wlog[HE][16];      // 8 KB zero-padded W_logit (8 -> 16 cols)
  const int tid  = threadIdx.x;
  const int wave = tid >> 5;
  const int lane = tid & 31;
  const size_t rowBase = (size_t)blockIdx.x * 64;

  // stage padded W_logit
  for (int idx = tid; idx < HE * 16; idx += 256) {
    const int kk = idx >> 4, cc = idx & 15;
    wlog[kk][cc] = (cc < 8) ? W_logit[kk * 8 + cc] : 0.f;
  }

  // LayerNorm: each wave handles 8 rows, 4 elements per lane
  float glv[4], blv[4];
  for (int j = 0; j < 4; ++j) { glv[j] = lg[lane + 32 * j]; blv[j] = lb[lane + 32 * j]; }
  for (int rr = 0; rr < 8; ++rr) {
    const int r = wave * 8 + rr;
    const float* src = edge_feat + (rowBase + r) * HE;
    float x[4]; float s = 0.f, s2 = 0.f;
    for (int j = 0; j < 4; ++j) {
      x[j] = src[lane + 32 * j];
      s += x[j]; s2 += x[j] * x[j];
    }
    for (int o = 16; o; o >>= 1) {
      s  += __shfl_xor(s,  o, 32);
      s2 += __shfl_xor(s2, o, 32);
    }
    const float mean = s * (1.f / HE);
    const float var  = s2 * (1.f / HE) - mean * mean;
    const float inv  = rsqrtf(var + 1e-5f);
    for (int j = 0; j < 4; ++j)
      ei[r][lane + 32 * j] = (x[j] - mean) * inv * glv[j] + blv[j];
  }
  __syncthreads();

  const int half = lane >> 4;
  const int l    = lane & 15;
  // LDS byte offset of the panel buffer (low 32 bits of flat shared address)
  const uint32_t wpan_base = (uint32_t)(uintptr_t)(&wpan[0][0]);

  // 8 accumulators per wave: tiles tt = wave + 8*i  (tileM = tt/16, tileN = tt%16)
  v8f acc[8] = {V8F_ZERO, V8F_ZERO, V8F_ZERO, V8F_ZERO,
                V8F_ZERO, V8F_ZERO, V8F_ZERO, V8F_ZERO};

  for (int kp = 0; kp < HE / 16; ++kp) {
    // ---- async-stage panel: W_edge[kp*16 .. kp*16+16)[0..256) = 16 KB contiguous
    const char* gsrc = (const char*)(W_edge + (size_t)kp * 16 * HN);
#pragma unroll
    for (int j = 0; j < 4; ++j) {
      const int idx = tid + j * 256;                 // 16-byte chunk index, 0..1023
      const uint32_t ldsa = wpan_base + (uint32_t)idx * 16u;
      const void* ga = gsrc + (size_t)idx * 16;
      asm volatile("global_load_async_to_lds_b128 %0, %1, off"
                   :: "v"(ldsa), "v"(ga) : "memory");
    }
    asm volatile("s_wait_asynccnt 0x0" ::: "memory");
    __syncthreads();                                 // panel visible to all waves

    // ---- 4 WMMA steps per tile from LDS
#pragma unroll
    for (int i = 0; i < 8; ++i) {
      const int tt    = wave + 8 * i;
      const int tileM = tt >> 4;
      const int tileN = tt & 15;
      const int arow  = tileM * 16 + l;
      const int bcol  = tileN * 16 + l;
#pragma unroll
      for (int k4 = 0; k4 < 4; ++k4) {
        const int kl = k4 * 4 + half * 2;            // k within panel
        const int kg = kp * 16 + kl;                 // k within ei
        v2f a = *(const v2f*)(&ei[arow][kg]);        // ds_load_b64
        v2f b;
        b.x = wpan[kl][bcol];
        b.y = wpan[kl + 1][bcol];
        acc[i] = __builtin_amdgcn_wmma_f32_16x16x4_f32(false, a, false, b,
                                                       (short)0, acc[i], false, false);
      }
    }
    __syncthreads();                                 // done with panel before overwrite
  }

  // ---- epilogue: bias + mask, store edge_new
#pragma unroll
  for (int i = 0; i < 8; ++i) {
    const int tt    = wave + 8 * i;
    const int tileM = tt >> 4;
    const int tileN = tt & 15;
    const int cc = tileN * 16 + l;
    const float bv = b_edge[cc];
    for (int r0 = 0; r0 < 8; ++r0) {
      const size_t gr = rowBase + tileM * 16 + (half ? 8 + r0 : r0);
      edge_new[gr * HN + cc] = (acc[i][r0] + bv) * mask[gr];
    }
  }

  // ---- logit GEMM: 4 tiles of 16x16 (cols 8..15 zero pad), waves 0..3
  if (wave < 4) {
    const int tileM = wave;
    const int arow  = tileM * 16 + l;
    v8f lacc = V8F_ZERO;
    for (int k = 0; k < HE; k += 4) {
      const int kb = k + half * 2;
      v2f a = *(const v2f*)(&ei[arow][kb]);
      v2f b;
      b.x = wlog[kb][l];
      b.y = wlog[kb + 1][l];
      lacc = __builtin_amdgcn_wmma_f32_16x16x4_f32(false, a, false, b,
                                                   (short)0, lacc, false, false);
    }
    if (l < 8) {
      const float bv = b_logit[l];
      for (int r0 = 0; r0 < 8; ++r0) {
        const size_t gr = rowBase + tileM * 16 + (half ? 8 + r0 : r0);
        const float val = lacc[r0] + bv;
        // mask==0 => logit = NEG => sigmoid = 0 exactly
        sig_out[gr * 8 + l] = (mask[gr] > 0.f) ? (1.f / (1.f + expf(-val))) : 0.f;
      }
    }
  }
}

// ============================================================
// Kernel 4: attention aggregation + exact GELU.  one block per (b,n)
// attn(b,n,:) = sum_m (edge_new[e]+ctx[m]) * sig[e,h] / max(sum sig, 1e-5)
// ============================================================
__global__ __launch_bounds__(256)
void attn_gelu_kernel(const float* __restrict__ edge_new, const float* __restrict__ sig,
                      const float* __restrict__ node_new, float* __restrict__ comb) {
  __shared__ float ssig[(NN - 1) * NH];   // 191*8 sigmoids
  __shared__ int   seidx[NN - 1];
  __shared__ float ssum[NH];
  const int tid = threadIdx.x;
  const int bn  = blockIdx.x;
  const int b   = bn / NN;
  const int n   = bn - b * NN;

  if (tid < NN - 1) {
    const int m = (tid < n) ? tid : tid + 1;
    const int i = (n < m) ? n : m;
    const int j = (n < m) ? m : n;
    const int e = i * (NN - 1) - (i * (i - 1)) / 2 + (j - i - 1);
    seidx[tid] = e;
    const float* sp = sig + ((size_t)b * EE + e) * NH;
    for (int h = 0; h < NH; ++h) ssig[tid * NH + h] = sp[h];
  }
  __syncthreads();
  if (tid < NH) {
    float s = 0.f;
    for (int t = 0; t < NN - 1; ++t) s += ssig[t * NH + tid];
    ssum[tid] = fmaxf(s, 1e-5f);
  }
  __syncthreads();

  const int h = tid >> 5;                  // head = col / DH
  const float* enb = edge_new + (size_t)b * EE * HN;
  const float* ctx = node_new + (size_t)b * NN * (2 * HN) + HN;   // ctx half
  float acc = 0.f;
  for (int t = 0; t < NN - 1; ++t) {
    const int e = seidx[t];
    const int m = (t < n) ? t : t + 1;
    const float w = ssig[t * NH + h];
    acc += (enb[(size_t)e * HN + tid] + ctx[(size_t)m * (2 * HN) + tid]) * w;
  }
  const float attn = acc / ssum[h];
  const float x = node_new[((size_t)b * NN + n) * (2 * HN) + tid] + attn; // self half
  const float gelu = 0.5f * x * (1.f + erff(x * 0.70710678118654752f));
  comb[(size_t)bn * HN + tid] = gelu;
}

// ============================================================
// Kernel 5: gated residual output
// ============================================================
__global__ __launch_bounds__(256)
void gate_out_kernel(const float* __restrict__ skipv, const float* __restrict__ node_feat,
                     float* __restrict__ out) {
  const int idx = blockIdx.x * 256 + threadIdx.x;     // 0 .. 196607
  const int row = idx >> 8;
  const int c   = idx & (HN - 1);
  const float val = skipv[(size_t)row * (2 * HN) + c];
  const float glg = skipv[(size_t)row * (2 * HN) + HN + c];
  const float g   = 1.f / (1.f + expf(-glg));
  const float nf  = node_feat[idx];
  out[idx] = nf * (1.f - g) + val * g;
}

// ============================================================
extern "C" void kernel_launch(void* const* d_in, const int* in_sizes, int n_in,
                              void* d_out, int out_size, void* d_ws, size_t ws_size,
                              hipStream_t stream) {
  const float* node_feat = (const float*)d_in[0];
  const float* edge_feat = (const float*)d_in[1];
  const float* mask_val  = (const float*)d_in[2];
  const float* ln_node_g = (const float*)d_in[3];
  const float* ln_node_b = (const float*)d_in[4];
  const float* ln_edge_g = (const float*)d_in[5];
  const float* ln_edge_b = (const float*)d_in[6];
  const float* W_node    = (const float*)d_in[7];
  const float* b_node    = (const float*)d_in[8];
  const float* W_edge    = (const float*)d_in[9];
  const float* b_edge    = (const float*)d_in[10];
  const float* W_logit   = (const float*)d_in[11];
  const float* b_logit   = (const float*)d_in[12];
  const float* W_skip    = (const float*)d_in[13];
  const float* b_skip    = (const float*)d_in[14];
  // x_idx1 / x_idx2 (d_in[15], d_in[16]) are the deterministic triu indices;
  // we regenerate them analytically on device.

  float* ws = (float*)d_ws;
  float* node_norm = ws;                                   // 768*256
  float* node_new  = node_norm + (size_t)ROWS_NODE * HN;   // 768*512
  float* combv     = node_new + (size_t)ROWS_NODE * 2 * HN;// 768*256
  float* skipv     = combv + (size_t)ROWS_NODE * HN;       // 768*512
  float* sigv      = skipv + (size_t)ROWS_NODE * 2 * HN;   // B*E*8
  float* edge_new  = sigv + (size_t)ROWS_EDGE * NH;        // B*E*256

  // 1) node LN
  node_ln_kernel<<<ROWS_NODE, 256, 0, stream>>>(node_feat, ln_node_g, ln_node_b, node_norm);

  // 2) node GEMM: (768,256)@(256,512): 48*32=1536 tiles / 8 waves = 192 blocks
  gemm_f32_wmma<<<192, 256, 0, stream>>>(node_norm, W_node, b_node, node_new,
                                         ROWS_NODE, HN, 2 * HN);

  // 3) fused edge LN + edge/logit GEMMs: 73344 rows / 64 = 1146 blocks
  edge_ln_gemm<<<ROWS_EDGE / 64, 256, 0, stream>>>(edge_feat, mask_val,
                                                   ln_edge_g, ln_edge_b,
                                                   W_edge, b_edge, W_logit, b_logit,
                                                   edge_new, sigv);

  // 4) attention + GELU: one block per (b,n)
  attn_gelu_kernel<<<ROWS_NODE, 256, 0, stream>>>(edge_new, sigv, node_new, combv);

  // 5) skip GEMM: (768,256)@(256,512)
  gemm_f32_wmma<<<192, 256, 0, stream>>>(combv, W_skip, b_skip, skipv,
                                         ROWS_NODE, HN, 2 * HN);

  // 6) gated residual
  gate_out_kernel<<<ROWS_NODE, 256, 0, stream>>>(skipv, node_feat, (float*)d_out);
}